// ChamferDistanceBlock_24240795418787
// MI455X (gfx1250) — compile-verified
//
#include <hip/hip_runtime.h>
#include <hip/hip_bf16.h>
#include <math.h>

typedef __attribute__((ext_vector_type(2))) float v2f;
typedef __attribute__((ext_vector_type(8))) float v8f;

#define TILE_M 128   // rays per block (8 waves * 16 rows)

#if defined(__gfx1250__) && \
    __has_builtin(__builtin_amdgcn_global_load_async_to_lds_b32) && \
    __has_builtin(__builtin_amdgcn_s_wait_asynccnt)
#define HAVE_ASYNC_LDS 1
#else
#define HAVE_ASYNC_LDS 0
#endif

#if HAVE_ASYNC_LDS
typedef __attribute__((address_space(1))) int gas_int;   // global
typedef __attribute__((address_space(3))) int las_int;   // LDS
#endif

// shared-memory region-0 size in floats: max(raw pc staging, 8x16x16 min tile),
// rounded to 16B. MUST match the formula used in kernel_launch.
__host__ __device__ static inline int region0_floats(int N) {
    int r = N * 3;
    if (r < 8 * 16 * 16) r = 8 * 16 * 16;
    return (r + 3) & ~3;
}

// ---------------------------------------------------------------------------
// Kernel 1: per-batch max_distance = max_n ||cam_loc - pc_n||, zero accums.
// ---------------------------------------------------------------------------
__global__ __launch_bounds__(256)
void chamfer_prep(const float* __restrict__ c, const float* __restrict__ pc,
                  float* __restrict__ maxdist, float* __restrict__ accum, int N) {
    const int b = blockIdx.x;
    const float* cb = c + b * 25;
    const float tx = cb[3], ty = cb[7], tz = cb[11];
    const float* pcb = pc + (size_t)b * N * 3;

    float mx = 0.0f;
    for (int n = threadIdx.x; n < N; n += blockDim.x) {
        float dx = tx - pcb[n * 3 + 0];
        float dy = ty - pcb[n * 3 + 1];
        float dz = tz - pcb[n * 3 + 2];
        mx = fmaxf(mx, sqrtf(dx * dx + dy * dy + dz * dz));
    }
    __shared__ float red[256];
    red[threadIdx.x] = mx;
    __syncthreads();
    for (int s = 128; s > 0; s >>= 1) {
        if (threadIdx.x < s) red[threadIdx.x] = fmaxf(red[threadIdx.x], red[threadIdx.x + s]);
        __syncthreads();
    }
    if (threadIdx.x == 0) {
        maxdist[b]       = red[0];
        accum[b * 2 + 0] = 0.0f;
        accum[b * 2 + 1] = 0.0f;
    }
}

// ---------------------------------------------------------------------------
// Kernel 2: main WMMA chamfer. Grid = (M/TILE_M, B), block = 256 (8 waves).
// LDS layout: [region0: raw pc staging, later reused as 8x16x16 min tile]
//             [frag: N16 x 4 floats = B-matrix fragments]
// ---------------------------------------------------------------------------
__global__ __launch_bounds__(256)
void chamfer_main(const float* __restrict__ c, const float* __restrict__ depth,
                  const float* __restrict__ pc, const float* __restrict__ maxdist,
                  float* __restrict__ accum, int res, int N) {
    extern __shared__ float smem[];
    const int  M    = res * res;
    const int  N16  = (N + 15) & ~15;
    const int  b    = blockIdx.y;
    const int  lane = threadIdx.x & 31;
    const int  wave = threadIdx.x >> 5;
    const int  nlo  = lane & 15;
    const int  kh   = lane >> 4;

    float* lds_frag = smem + region0_floats(N);
    const float* pcb = pc + (size_t)b * N * 3;

    // ---- stage raw pc into LDS (async DMA path on CDNA5) -------------------
#if HAVE_ASYNC_LDS
    {
        float* lds_raw = smem;
        for (int idx = threadIdx.x; idx < N * 3; idx += 256) {
            __builtin_amdgcn_global_load_async_to_lds_b32(
                (gas_int*)(pcb + idx), (las_int*)(lds_raw + idx), 0, 0);
        }
        __builtin_amdgcn_s_wait_asynccnt(0);
    }
    __syncthreads();
    const float* rawsrc = smem;      // read staged copy from LDS
#else
    const float* rawsrc = pcb;       // fallback: read global directly
#endif

    // ---- build B-matrix fragments: frag[n] = [-2qx, -2qy, -2qz, ||q||^2] ---
    for (int n = threadIdx.x; n < N16; n += 256) {
        float qx = 0.f, qy = 0.f, qz = 0.f, n2 = 3.0e38f;  // pad col -> huge
        if (n < N) {
            qx = rawsrc[n * 3 + 0];
            qy = rawsrc[n * 3 + 1];
            qz = rawsrc[n * 3 + 2];
            n2 = qx * qx + qy * qy + qz * qz;
        }
        lds_frag[n * 4 + 0] = -2.0f * qx;
        lds_frag[n * 4 + 1] = -2.0f * qy;
        lds_frag[n * 4 + 2] = -2.0f * qz;
        lds_frag[n * 4 + 3] = n2;
    }
    __syncthreads();

    // ---- per-lane ray lift (lanes l and l+16 duplicate ray l) --------------
    const float* cb = c + b * 25;
    const float r00 = cb[0], r01 = cb[1], r02 = cb[2],  tx = cb[3];
    const float r10 = cb[4], r11 = cb[5], r12 = cb[6],  ty = cb[7];
    const float r20 = cb[8], r21 = cb[9], r22 = cb[10], tz = cb[11];
    const float fx = cb[16], sk = cb[17], cx = cb[18];
    const float fy = cb[20], cy = cb[21];

    const int m  = blockIdx.x * TILE_M + wave * 16 + nlo;
    const int mc = (m < M) ? m : (M - 1);
    const int   i = mc / res, j = mc % res;
    const float x = (j + 0.5f) / (float)res;
    const float y = (i + 0.5f) / (float)res;
    const float xl = (x - cx + cy * sk / fy - sk * y / fy) / fx;
    const float yl = (y - cy) / fy;
    const float wxv = r00 * xl + r01 * yl + r02 + tx;
    const float wyv = r10 * xl + r11 * yl + r12 + ty;
    const float wzv = r20 * xl + r21 * yl + r22 + tz;
    float dx = wxv - tx, dy = wyv - ty, dz = wzv - tz;
    const float nrm = fmaxf(sqrtf(dx * dx + dy * dy + dz * dz), 1e-12f);
    dx /= nrm; dy /= nrm; dz /= nrm;

    const float dpt = depth[(size_t)b * M + mc];
    const float px = dpt * dx + tx;
    const float py = dpt * dy + ty;
    const float pz = dpt * dz + tz;
    const float p2 = px * px + py * py + pz * pz;
    const float msk = ((dpt < maxdist[b]) && (m < M)) ? 1.0f : 0.0f;

    // A-fragment: row m = [px, py, pz, 1]; lane holds K = 2*kh, 2*kh+1
    v2f a;
    a.x = kh ? pz   : px;
    a.y = kh ? 1.0f : py;

    // ---- hot loop: 1 ds_load_b64 + 1 v_wmma_f32_16x16x4_f32 + 8 v_min ------
    v8f runmin;
#pragma unroll
    for (int r = 0; r < 8; ++r) runmin[r] = 3.0e38f;

    const float* fp = lds_frag + nlo * 4 + kh * 2;   // 8B-aligned, conflict-free
    const v8f zero = {};
#pragma unroll 4
    for (int t = 0; t < N16; t += 16) {
        v2f bb = *(const v2f*)(fp + t * 4);
        v8f s = __builtin_amdgcn_wmma_f32_16x16x4_f32(
            false, a, false, bb, (short)0, zero, false, false);
#pragma unroll
        for (int r = 0; r < 8; ++r) runmin[r] = fminf(runmin[r], s[r]);
    }

    // ---- per-row min over 16 residual columns via LDS transpose ------------
    // C/D layout: VGPR r -> lanes 0-15 row r, lanes 16-31 row r+8.
    __syncthreads();                                  // frag no longer needed
    float (*mint)[16][16] = (float (*)[16][16])smem;  // reuse region0
#pragma unroll
    for (int r = 0; r < 8; ++r) {
        mint[wave][r + (kh ? 8 : 0)][nlo] = runmin[r];
    }
    __syncthreads();

    float contrib = 0.0f, cnt = 0.0f;
    if (kh == 0) {
        float v = mint[wave][nlo][0];
#pragma unroll
        for (int col = 1; col < 16; ++col) v = fminf(v, mint[wave][nlo][col]);
        const float min_d2 = fmaxf(v + p2, 0.0f);
        contrib = min_d2 * msk;
        cnt     = msk;
    }
#pragma unroll
    for (int off = 16; off >= 1; off >>= 1) {
        contrib += __shfl_xor(contrib, off, 32);
        cnt     += __shfl_xor(cnt,     off, 32);
    }
    if (lane == 0) {
        atomicAdd(&accum[b * 2 + 0], contrib);
        atomicAdd(&accum[b * 2 + 1], cnt);
    }
}

// ---------------------------------------------------------------------------
// Kernel 3: loss = num / max(cnt, 1)
// ---------------------------------------------------------------------------
__global__ void chamfer_final(const float* __restrict__ accum,
                              float* __restrict__ out, int B) {
    const int b = threadIdx.x;
    if (b < B) out[b] = accum[b * 2 + 0] / fmaxf(accum[b * 2 + 1], 1.0f);
}

// ---------------------------------------------------------------------------
extern "C" void kernel_launch(void* const* d_in, const int* in_sizes, int n_in,
                              void* d_out, int out_size, void* d_ws, size_t ws_size,
                              hipStream_t stream) {
    const float* c     = (const float*)d_in[0];  // B x 25
    const float* depth = (const float*)d_in[1];  // B x res x res
    const float* pc    = (const float*)d_in[2];  // B x N x 3
    const int B = in_sizes[0] / 25;
    const int M = in_sizes[1] / B;
    const int res = (int)(sqrt((double)M) + 0.5);
    const int N = in_sizes[2] / (3 * B);
    const int N16 = (N + 15) & ~15;

    float* maxdist = (float*)d_ws;        // B floats
    float* accum   = maxdist + B;         // B x 2 floats
    float* out     = (float*)d_out;       // B floats

    const size_t shbytes =
        (size_t)region0_floats(N) * sizeof(float) +     // raw staging / min tile
        (size_t)N16 * 4 * sizeof(float);                // B fragments

    chamfer_prep<<<dim3(B), dim3(256), 0, stream>>>(c, pc, maxdist, accum, N);

    dim3 grid((M + TILE_M - 1) / TILE_M, B);
    chamfer_main<<<grid, dim3(256), shbytes, stream>>>(c, depth, pc, maxdist,
                                                       accum, res, N);

    chamfer_final<<<dim3(1), dim3(64), 0, stream>>>(accum, out, B);
}